// QuaternionLinear_68238440399557
// MI455X (gfx1250) — compile-verified
//
#include <hip/hip_runtime.h>

// ---------------------------------------------------------------------------
// Quaternion linear layer:  out[b,o,:] = sum_i  x[b,i,:] (Hamilton) w[o,i,:]  + bias[o,:]
// BATCH=4096, IN=1024, OUT=1024, all fp32.
//
// Memory-bound (144MB @ 23.3TB/s ~ 6.2us  vs  <1us of f32 matrix math), so we
// keep full fp32 precision and use V_WMMA_F32_16X16X4_F32.
// Weight (16MB) is L2-resident (192MB L2); x is streamed once.
// ---------------------------------------------------------------------------

typedef float v2f __attribute__((ext_vector_type(2)));
typedef float v8f __attribute__((ext_vector_type(8)));

#define Q_BATCH 4096
#define Q_IN    1024
#define Q_OUT   1024

// D = A(16x4 f32) * B(4x16 f32) + C(16x16 f32)
#define WMMA_F32(a, b, c) \
    __builtin_amdgcn_wmma_f32_16x16x4_f32(false, (a), false, (b), (short)0, (c), false, false)

// Hamilton-product accumulation: 16 WMMAs covering the 4x4 sign pattern.
//  out_r = ar*br - ai*bi - aj*bj - ak*bk
//  out_i = ar*bi + ai*br + aj*bk - ak*bj
//  out_j = ar*bj - ai*bk + aj*br + ak*bi
//  out_k = ar*bk + ai*bj - aj*bi + ak*br
#define QWMMA(br, bi, bj, bk, cr, ci, cj, ck)         \
    do {                                              \
        cr = WMMA_F32(ar,  br, cr);                   \
        cr = WMMA_F32(nai, bi, cr);                   \
        cr = WMMA_F32(naj, bj, cr);                   \
        cr = WMMA_F32(nak, bk, cr);                   \
        ci = WMMA_F32(ar,  bi, ci);                   \
        ci = WMMA_F32(ai,  br, ci);                   \
        ci = WMMA_F32(aj,  bk, ci);                   \
        ci = WMMA_F32(nak, bj, ci);                   \
        cj = WMMA_F32(ar,  bj, cj);                   \
        cj = WMMA_F32(nai, bk, cj);                   \
        cj = WMMA_F32(aj,  br, cj);                   \
        cj = WMMA_F32(ak,  bi, cj);                   \
        ck = WMMA_F32(ar,  bk, ck);                   \
        ck = WMMA_F32(ai,  bj, ck);                   \
        ck = WMMA_F32(naj, bi, ck);                   \
        ck = WMMA_F32(ak,  br, ck);                   \
    } while (0)

// Block = 256 threads = 8 waves (wave32). Waves tile 4 (batch) x 2 (out):
// block tile = 64 batch x 64 out; each wave owns 16 batch x 32 out.
__global__ __launch_bounds__(256)
void quat_linear_wmma_f32(const float* __restrict__ x,
                          const float* __restrict__ w,
                          const float* __restrict__ bias,
                          float* __restrict__ out)
{
    const int lane  = threadIdx.x & 31;
    const int wave  = threadIdx.x >> 5;
    const int waveM = wave >> 1;         // 0..3  -> batch sub-tile
    const int waveN = wave & 1;          // 0..1  -> out   sub-tile

    const int b0 = blockIdx.y * 64 + waveM * 16;   // batch tile origin
    const int o0 = blockIdx.x * 64 + waveN * 32;   // out   tile origin

    // A 16x4 f32 fragment layout (ISA 7.12.2): lanes 0-15 -> M=lane, K={0,1};
    // lanes 16-31 -> M=lane-16, K={2,3}.  VGPR0=K even slot, VGPR1=K odd slot.
    const int mr   = lane & 15;          // row within 16 (A: M, B: N, C/D: N)
    const int ksel = lane >> 4;          // 0: K pair {0,1} / C rows 0-7 ; 1: {2,3} / rows 8-15

    // Quaternion-granular pointers (each float4 = one quaternion).
    const float4* __restrict__ xrow  = (const float4*)x + (size_t)(b0 + mr)      * Q_IN;
    const float4* __restrict__ wrow0 = (const float4*)w + (size_t)(o0 + mr)      * Q_IN;
    const float4* __restrict__ wrow1 = (const float4*)w + (size_t)(o0 + 16 + mr) * Q_IN;

    // Accumulators: 4 quaternion components x 2 N-subtiles, each 16x16 f32.
    v8f cr0 = {}, ci0 = {}, cj0 = {}, ck0 = {};
    v8f cr1 = {}, ci1 = {}, cj1 = {}, ck1 = {};

#pragma unroll 2
    for (int k0 = 0; k0 < Q_IN; k0 += 4) {
        const int kq = k0 + 2 * ksel;    // this lane's first quaternion index

        // Prefetch 512B (32 quaternions) ahead on each stream (global_prefetch_b8).
        __builtin_prefetch(xrow  + kq + 32, 0, 3);
        __builtin_prefetch(wrow0 + kq + 32, 0, 3);
        __builtin_prefetch(wrow1 + kq + 32, 0, 3);

        // Two consecutive quaternions per lane: 2 x global_load_b128 each.
        const float4 a0 = xrow[kq],  a1 = xrow[kq + 1];
        const float4 p0 = wrow0[kq], p1 = wrow0[kq + 1];
        const float4 q0 = wrow1[kq], q1 = wrow1[kq + 1];

        // Component-split into WMMA f32 fragments (v2f = {K even, K odd}).
        const v2f ar = {a0.x, a1.x}, ai = {a0.y, a1.y};
        const v2f aj = {a0.z, a1.z}, ak = {a0.w, a1.w};
        const v2f nai = -ai, naj = -aj, nak = -ak;   // signs (f32 WMMA has no A/B neg)

        const v2f br0 = {p0.x, p1.x}, bi0 = {p0.y, p1.y};
        const v2f bj0 = {p0.z, p1.z}, bk0 = {p0.w, p1.w};
        const v2f br1 = {q0.x, q1.x}, bi1 = {q0.y, q1.y};
        const v2f bj1 = {q0.z, q1.z}, bk1 = {q0.w, q1.w};

        QWMMA(br0, bi0, bj0, bk0, cr0, ci0, cj0, ck0);
        QWMMA(br1, bi1, bj1, bk1, cr1, ci1, cj1, ck1);
    }

    // Epilogue: C/D layout (ISA 7.12.2): VGPR r holds M=r (lanes 0-15) and
    // M=r+8 (lanes 16-31), N = lane&15.  Fuse the 4 component planes + bias
    // into one 128-bit store per output quaternion.
    const float4* __restrict__ bias4 = (const float4*)bias;
    float4* __restrict__ out4 = (float4*)out;

    const int col0 = o0 + mr;
    const int col1 = o0 + 16 + mr;
    const float4 bv0 = bias4[col0];
    const float4 bv1 = bias4[col1];

#pragma unroll
    for (int r = 0; r < 8; ++r) {
        const int row = b0 + r + 8 * ksel;
        float4 o;
        o.x = cr0[r] + bv0.x;
        o.y = ci0[r] + bv0.y;
        o.z = cj0[r] + bv0.z;
        o.w = ck0[r] + bv0.w;
        out4[(size_t)row * Q_OUT + col0] = o;

        o.x = cr1[r] + bv1.x;
        o.y = ci1[r] + bv1.y;
        o.z = cj1[r] + bv1.z;
        o.w = ck1[r] + bv1.w;
        out4[(size_t)row * Q_OUT + col1] = o;
    }
}

extern "C" void kernel_launch(void* const* d_in, const int* in_sizes, int n_in,
                              void* d_out, int out_size, void* d_ws, size_t ws_size,
                              hipStream_t stream)
{
    const float* x    = (const float*)d_in[0];   // (4096, 1024, 4) fp32
    const float* w    = (const float*)d_in[1];   // (1024, 1024, 4) fp32
    const float* bias = (const float*)d_in[2];   // (1024, 4) fp32
    float* out        = (float*)d_out;           // (4096, 1024, 4) fp32

    dim3 grid(Q_OUT / 64, Q_BATCH / 64);         // (16, 64) blocks of 64x64 tiles
    dim3 block(256);                             // 8 waves (wave32)
    quat_linear_wmma_f32<<<grid, block, 0, stream>>>(x, w, bias, out);
}